// LsqQuantizerWeight_iterative_freezing_46729244180908
// MI455X (gfx1250) — compile-verified
//
#include <hip/hip_runtime.h>
#include <stdint.h>

// ---- problem constants (from reference) ----
#define ROWS 4096
#define COLS 11008
#define NTOTC 45088768LL        // 4096*11008
#define CHUNK 1024              // elements per block tile (per stream)
#define NCHUNK 11               // ceil(11008/1024); last tile = 768
#define EPSV 1e-5f
#define MOM 0.01f
#define FRZ_TH 0.01f

// ---- raw vector types ----
typedef float f32x4 __attribute__((ext_vector_type(4)));
typedef unsigned int u32x4 __attribute__((ext_vector_type(4)));
typedef int i32x8 __attribute__((ext_vector_type(8)));
typedef int i32x4 __attribute__((ext_vector_type(4)));

#if __has_builtin(__builtin_amdgcn_tensor_load_to_lds)
#define HAVE_TDM 1
#else
#define HAVE_TDM 0
#endif

#if HAVE_TDM
// Issue one 1-D TDM tile load: tile_elems f32 values from gsrc into LDS at byte address lds_byte_off.
// D# layout per CDNA5 ISA §8 (group0 128b, group1 256b, groups 2/3 zero/unused).
__device__ __forceinline__ void tdm_load_f32_tile(uint32_t lds_byte_off,
                                                  const float* gsrc,
                                                  int tile_elems) {
  uint64_t ga = (uint64_t)(uintptr_t)gsrc;
  u32x4 g0;
  g0[0] = 1u;                                              // count=1, user descriptor
  g0[1] = lds_byte_off;                                    // lds_addr
  g0[2] = (uint32_t)ga;                                    // global_addr[31:0]
  g0[3] = ((uint32_t)(ga >> 32) & 0x01FFFFFFu) | (2u << 30); // global_addr[56:32] | type=2
  i32x8 g1;
  g1[0] = (int)(2u << 16);                                 // data_size=4B; no multicast/iterate/pad
  g1[1] = (int)(((uint32_t)COLS & 0xFFFFu) << 16);         // tensor_dim0[15:0]=11008
  g1[2] = (int)((1u << 16) | ((uint32_t)COLS >> 16));      // tensor_dim1=1 | tensor_dim0[31:16]
  g1[3] = (int)(((uint32_t)tile_elems & 0xFFFFu) << 16);   // tile_dim0=tile_elems
  g1[4] = 0;                                               // tile_dim1=0 (1-D), tile_dim2=0
  g1[5] = (int)(uint32_t)COLS;                             // tensor_dim0_stride=11008
  g1[6] = 0;
  g1[7] = 0;
  i32x4 z4 = {0, 0, 0, 0};
#if defined(__clang_major__) && (__clang_major__ >= 23)
  i32x8 z8 = {0, 0, 0, 0, 0, 0, 0, 0};
  __builtin_amdgcn_tensor_load_to_lds(g0, g1, z4, z4, z8, 0);
#else
  __builtin_amdgcn_tensor_load_to_lds(g0, g1, z4, z4, 0);
#endif
}
#endif // HAVE_TDM

union F4 { f32x4 v; float a[4]; };

__global__ __launch_bounds__(256)
void lsq_freeze_main(const float* __restrict__ x,
                     const float* __restrict__ s,
                     const float* __restrict__ prev_x_int,
                     const float* __restrict__ prev_switch_dir,
                     const float* __restrict__ ema_osc,
                     const unsigned char* __restrict__ frozen,
                     const float* __restrict__ frozen_x_int,
                     const float* __restrict__ ema_x_int,
                     float* __restrict__ ob,
                     unsigned int* __restrict__ osc_acc)
{
  __shared__ float lx[CHUNK];
  __shared__ float lpx[CHUNK];
  __shared__ float lsd[CHUNK];
  __shared__ float leo[CHUNK];
  __shared__ float lfx[CHUNK];
  __shared__ float lex[CHUNK];
  __shared__ int red[256];

  const int row  = blockIdx.y;
  const int col0 = (int)blockIdx.x * CHUNK;
  const int len  = min(CHUNK, COLS - col0);
  const long long base = (long long)row * COLS + col0;

#if HAVE_TDM
  // TDM ignores EXEC, and tensor ops from the SAME wave serialize in the DMA
  // engine. Give each of waves 0..5 one stream so the 6 tile loads proceed
  // concurrently; each wave waits only on its own TENSORcnt, then barrier.
  {
    const int wave = (int)threadIdx.x >> 5;
    if (wave < 6) {
      switch (wave) {
        case 0: tdm_load_f32_tile((uint32_t)(uintptr_t)&lx[0],  x               + base, len); break;
        case 1: tdm_load_f32_tile((uint32_t)(uintptr_t)&lpx[0], prev_x_int      + base, len); break;
        case 2: tdm_load_f32_tile((uint32_t)(uintptr_t)&lsd[0], prev_switch_dir + base, len); break;
        case 3: tdm_load_f32_tile((uint32_t)(uintptr_t)&leo[0], ema_osc         + base, len); break;
        case 4: tdm_load_f32_tile((uint32_t)(uintptr_t)&lfx[0], frozen_x_int    + base, len); break;
        case 5: tdm_load_f32_tile((uint32_t)(uintptr_t)&lex[0], ema_x_int       + base, len); break;
      }
      __builtin_amdgcn_s_wait_tensorcnt(0);
    }
  }
  __syncthreads();
#else
  // Fallback: cooperative vectorized fill of LDS.
  for (int i = (int)threadIdx.x * 4; i < len; i += 256 * 4) {
    *(f32x4*)&lx[i]  = *(const f32x4*)&x[base + i];
    *(f32x4*)&lpx[i] = *(const f32x4*)&prev_x_int[base + i];
    *(f32x4*)&lsd[i] = *(const f32x4*)&prev_switch_dir[base + i];
    *(f32x4*)&leo[i] = *(const f32x4*)&ema_osc[base + i];
    *(f32x4*)&lfx[i] = *(const f32x4*)&frozen_x_int[base + i];
    *(f32x4*)&lex[i] = *(const f32x4*)&ema_x_int[base + i];
  }
  __syncthreads();
#endif

  const float ss = fmaxf(s[row], EPSV);   // forward value of s_scale

  int cnt = 0;
  const int t4 = (int)threadIdx.x * 4;
  if (t4 < len) {
    const long long g = base + t4;

    F4 vx, vpx, vsd, veo, vfx, vex;
    vx.v  = *(const f32x4*)&lx[t4];
    vpx.v = *(const f32x4*)&lpx[t4];
    vsd.v = *(const f32x4*)&lsd[t4];
    veo.v = *(const f32x4*)&leo[t4];
    vfx.v = *(const f32x4*)&lfx[t4];
    vex.v = *(const f32x4*)&lex[t4];
    // read-once byte stream: non-temporal dword load, unpack 4 bools
    const unsigned fzw = __builtin_nontemporal_load((const unsigned*)(frozen + g));

    F4 oOut, oPx, oSd, oEo, oFz, oFx, oEx;

#pragma unroll
    for (int c = 0; c < 4; ++c) {
      float xs = vx.a[c] / ss;
      float xc = fminf(fmaxf(xs, -8.0f), 7.0f);
      float r  = __builtin_rintf(xc);                 // round-half-even == jnp.round
      bool  fz = ((fzw >> (8 * c)) & 0xFFu) != 0u;
      float xi = fz ? vfx.a[c] : r;                   // x_int forward (with frozen subst)
      float delta = __builtin_rintf(vpx.a[c] - xi);
      float sd = (delta > 0.0f) ? 1.0f : ((delta < 0.0f) ? -1.0f : 0.0f);
      bool  sw = (delta != 0.0f);
      bool  oscb = (vsd.a[c] * sd == -1.0f);
      float osc = oscb ? 1.0f : 0.0f;
      float eo  = MOM * osc + (1.0f - MOM) * veo.a[c];
      bool  fw  = (eo > FRZ_TH);

      cnt += oscb ? 1 : 0;
      oOut.a[c] = xi * ss;
      oPx.a[c]  = xi;
      oSd.a[c]  = sw ? sd : vsd.a[c];
      oEo.a[c]  = eo;
      oFz.a[c]  = (fz || fw) ? 1.0f : 0.0f;
      oFx.a[c]  = fw ? __builtin_rintf(vex.a[c]) : vfx.a[c];
      oEx.a[c]  = MOM * xi + (1.0f - MOM) * vex.a[c];
    }

    // Output layout: [out | prev_x_int | prev_switch_dir | ema_osc | SUM(1) |
    //                 frozen | frozen_x_int | ema_x_int]
    // Write-once streams (2.4 GB total >> 192 MB L2): non-temporal stores.
    __builtin_nontemporal_store(oOut.v, (f32x4*)(ob + g));             // 16B aligned
    __builtin_nontemporal_store(oPx.v,  (f32x4*)(ob + NTOTC + g));
    __builtin_nontemporal_store(oSd.v,  (f32x4*)(ob + 2 * NTOTC + g));
    __builtin_nontemporal_store(oEo.v,  (f32x4*)(ob + 3 * NTOTC + g));
    // sections after the scalar are shifted by +1 float -> only 4B aligned
    float* p4 = ob + 4 * NTOTC + 1 + g;
    float* p5 = ob + 5 * NTOTC + 1 + g;
    float* p6 = ob + 6 * NTOTC + 1 + g;
#pragma unroll
    for (int c = 0; c < 4; ++c) {
      __builtin_nontemporal_store(oFz.a[c], p4 + c);
      __builtin_nontemporal_store(oFx.a[c], p5 + c);
      __builtin_nontemporal_store(oEx.a[c], p6 + c);
    }
  }

  // exact, deterministic reduction: integer tree in LDS + one u32 atomic per block
  red[threadIdx.x] = cnt;
  __syncthreads();
#pragma unroll
  for (int off = 128; off > 0; off >>= 1) {
    if ((int)threadIdx.x < off) red[threadIdx.x] += red[threadIdx.x + off];
    __syncthreads();
  }
  if (threadIdx.x == 0 && red[0] > 0) atomicAdd(osc_acc, (unsigned int)red[0]);
}

__global__ void lsq_init_acc(unsigned int* acc) {
  if (threadIdx.x == 0 && blockIdx.x == 0) *acc = 0u;
}

__global__ void lsq_fini_acc(const unsigned int* acc, float* dst) {
  if (threadIdx.x == 0 && blockIdx.x == 0) *dst = (float)(*acc);
}

extern "C" void kernel_launch(void* const* d_in, const int* in_sizes, int n_in,
                              void* d_out, int out_size, void* d_ws, size_t ws_size,
                              hipStream_t stream) {
  const float*         x   = (const float*)d_in[0];
  const float*         s   = (const float*)d_in[1];
  const float*         pxi = (const float*)d_in[2];
  const float*         psd = (const float*)d_in[3];
  const float*         emo = (const float*)d_in[4];
  const unsigned char* fz  = (const unsigned char*)d_in[5];   // numpy bool = 1 byte
  const float*         fxi = (const float*)d_in[6];
  const float*         exi = (const float*)d_in[7];
  float*        out = (float*)d_out;
  unsigned int* acc = (unsigned int*)d_ws;

  lsq_init_acc<<<1, 64, 0, stream>>>(acc);

  dim3 grid(NCHUNK, ROWS);   // 11 x 4096 = 45,056 blocks
  lsq_freeze_main<<<grid, 256, 0, stream>>>(x, s, pxi, psd, emo, fz, fxi, exi, out, acc);

  lsq_fini_acc<<<1, 64, 0, stream>>>(acc, out + 4 * NTOTC);
}